// BulkSpaceGenerator_78176994721884
// MI455X (gfx1250) — compile-verified
//
#include <hip/hip_runtime.h>

// ---- problem constants (match reference) ----
#define D_MODEL 1024
#define BULK_K  10
#define WIDE    (D_MODEL * BULK_K)   // 10240
#define N_SEQ   4096
#define NB      2
#define NROWS   (NB * N_SEQ)         // 8192

// scan decomposition
#define NCHUNK  16
#define CHLEN   (N_SEQ / NCHUNK)     // 256

// LDS row stride in floats, padded (+4) so A-fragment reads are bank-conflict free
#define LDS_STRIDE 1028

typedef __attribute__((ext_vector_type(2))) float v2f;
typedef __attribute__((ext_vector_type(8))) float v8f;

// ---------------------------------------------------------------------------
// Kernel 1: fused  t[g,d] = sum_k |(Xd @ W1)[g, k*1024+d] (+ b1 on first row)| / (k+1)
// where Xd[n] = x[n] - x[n-1] (x[-1] == 0 per batch).
// Block: 256 threads = 8 waves. Block owns one 16-row M tile (staged in LDS),
// each wave owns one 16-col d tile and accumulates all 10 k-slices at once.
// ---------------------------------------------------------------------------
__global__ __launch_bounds__(256) void bulk_gemm_kernel(
    const float* __restrict__ X,    // (8192, 1024)
    const float* __restrict__ W1,   // (1024, 10240) row-major
    const float* __restrict__ b1,   // (10240,)
    float* __restrict__ t)          // (8192, 1024)
{
  __shared__ float ldsA[16 * LDS_STRIDE];

  const int tid  = threadIdx.x;
  const int wave = tid >> 5;
  const int lane = tid & 31;
  const int half = lane >> 4;     // 0: lanes 0-15, 1: lanes 16-31
  const int ln   = lane & 15;

  const int m0 = blockIdx.y * 16;                    // global row tile base (0..8176)
  const int d0 = (blockIdx.x * 8 + wave) * 16;       // d column tile base for this wave
  const int col = d0 + ln;                           // this lane's output column

  // ---- stage differenced X tile (16 x 1024 floats) into LDS -------------
  // thread tid handles float4 column tid for all 16 rows.
  {
    const int c4 = tid;  // 0..255
    for (int r = 0; r < 16; ++r) {
      const int g = m0 + r;
      const int n = g & (N_SEQ - 1);
      float4 cur = ((const float4*)(X + (size_t)g * D_MODEL))[c4];
      if (n != 0) {
        const float4 prev = ((const float4*)(X + (size_t)(g - 1) * D_MODEL))[c4];
        cur.x -= prev.x; cur.y -= prev.y; cur.z -= prev.z; cur.w -= prev.w;
      }
      *(float4*)(&ldsA[r * LDS_STRIDE + c4 * 4]) = cur;
    }
  }
  __syncthreads();

  // ---- 10 simultaneous 16x16 fp32 GEMMs over the K=1024 reduction -------
  v8f acc[BULK_K];
  {
    v8f z = {0.f, 0.f, 0.f, 0.f, 0.f, 0.f, 0.f, 0.f};
#pragma unroll
    for (int k = 0; k < BULK_K; ++k) acc[k] = z;
  }

  for (int kk = 0; kk < D_MODEL; kk += 4) {
    // A 16x4 fragment (ISA 7.12.2): lanes 0-15 hold K=kk,kk+1; lanes 16-31 hold K=kk+2,kk+3
    v2f a = *(const v2f*)(&ldsA[ln * LDS_STRIDE + kk + 2 * half]);

    const float* wrow = W1 + (size_t)(kk + 2 * half) * WIDE + col;
    if (kk + 4 < D_MODEL)
      __builtin_prefetch(W1 + (size_t)(kk + 4) * WIDE + col, 0, 1);

#pragma unroll
    for (int k = 0; k < BULK_K; ++k) {
      // B 4x16 fragment (symmetric layout): lanes 0-15 hold rows kk,kk+1; 16-31 rows kk+2,kk+3
      v2f b;
      b.x = wrow[(size_t)k * D_MODEL];
      b.y = wrow[(size_t)k * D_MODEL + WIDE];
      acc[k] = __builtin_amdgcn_wmma_f32_16x16x4_f32(
          false, a, false, b, (short)0, acc[k], false, false);
    }
  }

  // ---- epilogue: t = sum_k |y (+ b1 on first row of batch)| / (k+1) -----
  float tout[8];
#pragma unroll
  for (int i = 0; i < 8; ++i) tout[i] = 0.f;

  const bool firstRowTile = ((m0 & (N_SEQ - 1)) == 0);
#pragma unroll
  for (int k = 0; k < BULK_K; ++k) {
    const float coef = 1.0f / (float)(k + 1);
    v8f y = acc[k];
    // C/D layout: VGPR i, lanes 0-15 -> M=i ; lanes 16-31 -> M=8+i.
    // Batch-first row (M==0) is i==0 on the low half-wave: bias survives there.
    if (firstRowTile && half == 0) y[0] += b1[k * D_MODEL + col];
#pragma unroll
    for (int i = 0; i < 8; ++i) tout[i] += coef * __builtin_fabsf(y[i]);
  }

  float* trow = t + (size_t)(m0 + 8 * half) * D_MODEL + col;
#pragma unroll
  for (int i = 0; i < 8; ++i) trow[(size_t)i * D_MODEL] = tout[i];
}

// ---------------------------------------------------------------------------
// Kernel 2a: per-chunk sums over n.  partial[b, chunk, d] = sum_{n in chunk} t
// grid (4, 16, 2), 256 threads; fully coalesced in d.
// ---------------------------------------------------------------------------
__global__ __launch_bounds__(256) void chunk_sum_kernel(
    const float* __restrict__ t, float* __restrict__ partial)
{
  const int d     = blockIdx.x * 256 + threadIdx.x;
  const int chunk = blockIdx.y;
  const int b     = blockIdx.z;
  const float* p = t + ((size_t)b * N_SEQ + (size_t)chunk * CHLEN) * D_MODEL + d;
  float s = 0.f;
  for (int n = 0; n < CHLEN; ++n) s += p[(size_t)n * D_MODEL];
  partial[((size_t)b * NCHUNK + chunk) * D_MODEL + d] = s;
}

// ---------------------------------------------------------------------------
// Kernel 2b: exclusive scan of chunk sums in place.  grid (4, 1, 2).
// ---------------------------------------------------------------------------
__global__ __launch_bounds__(256) void chunk_scan_kernel(float* __restrict__ partial)
{
  const int d = blockIdx.x * 256 + threadIdx.x;
  const int b = blockIdx.z;
  float run = 0.f;
  for (int c = 0; c < NCHUNK; ++c) {
    const size_t idx = ((size_t)b * NCHUNK + c) * D_MODEL + d;
    const float v = partial[idx];
    partial[idx] = run;
    run += v;
  }
}

// ---------------------------------------------------------------------------
// Kernel 2c: final cumsum within each chunk, seeded by the chunk offset.
// grid (4, 16, 2); fully coalesced in d.
// ---------------------------------------------------------------------------
__global__ __launch_bounds__(256) void cumsum_kernel(
    const float* __restrict__ t, const float* __restrict__ partial,
    float* __restrict__ out)
{
  const int d     = blockIdx.x * 256 + threadIdx.x;
  const int chunk = blockIdx.y;
  const int b     = blockIdx.z;
  float run = partial[((size_t)b * NCHUNK + chunk) * D_MODEL + d];
  const size_t base = ((size_t)b * N_SEQ + (size_t)chunk * CHLEN) * D_MODEL + d;
  for (int n = 0; n < CHLEN; ++n) {
    run += t[base + (size_t)n * D_MODEL];
    out[base + (size_t)n * D_MODEL] = run;
  }
}

// ---------------------------------------------------------------------------
extern "C" void kernel_launch(void* const* d_in, const int* in_sizes, int n_in,
                              void* d_out, int out_size, void* d_ws, size_t ws_size,
                              hipStream_t stream) {
  (void)in_sizes; (void)n_in; (void)out_size; (void)ws_size;
  const float* X  = (const float*)d_in[0];   // boundary_tokens (2,4096,1024)
  const float* W1 = (const float*)d_in[1];   // (1024, 10240)
  const float* b1 = (const float*)d_in[2];   // (10240,)
  float* out = (float*)d_out;                // (2,4096,1024)

  // workspace: t (8192*1024 f32 = 32 MiB) then partial (2*16*1024 f32)
  float* t       = (float*)d_ws;
  float* partial = t + (size_t)NROWS * D_MODEL;

  bulk_gemm_kernel<<<dim3(D_MODEL / 128, NROWS / 16, 1), 256, 0, stream>>>(X, W1, b1, t);
  chunk_sum_kernel<<<dim3(D_MODEL / 256, NCHUNK, NB), 256, 0, stream>>>(t, partial);
  chunk_scan_kernel<<<dim3(D_MODEL / 256, 1, NB), 256, 0, stream>>>(partial);
  cumsum_kernel<<<dim3(D_MODEL / 256, NCHUNK, NB), 256, 0, stream>>>(t, partial, out);
}